// CombinedLoss_29437705847569
// MI455X (gfx1250) — compile-verified
//
#include <hip/hip_runtime.h>
#include <cstddef>

typedef __attribute__((ext_vector_type(2))) float v2f;
typedef __attribute__((ext_vector_type(8))) float v8f;

// Problem constants (fixed by the reference setup_inputs)
constexpr int B  = 4;
constexpr int S  = 10;
constexpr int N  = 2048;
constexpr int D  = 128;
constexpr int BSN = B * S;            // 40 (b,s) slices

constexpr int WAVES_PER_WG  = 8;      // 256 threads, wave32
constexpr int ROWS_PER_WAVE = 16;
constexpr int ROWS_PER_WG   = WAVES_PER_WG * ROWS_PER_WAVE;  // 128
constexpr int ROW_TILES     = N / ROWS_PER_WG;               // 16
constexpr int COL_TILES     = N / 16;                        // 128

constexpr int KL_N   = B * S * D;         // 5120
constexpr int KL_BLK = (KL_N + 255) / 256;            // 20
constexpr int TMP_N  = B * (S - 1) * N;   // 73728
constexpr int TMP_BLK = (TMP_N + 255) / 256;          // 288
constexpr int RECON_PARTIALS = 2 * BSN * ROW_TILES;   // 1280

// ---------------------------------------------------------------------------
// Chamfer one-directional row-min kernel.
//  dir==0: rows = pred, cols = target  -> sum_i min_j d
//  dir==1: rows = target, cols = pred  -> sum_j min_i d   (d symmetric)
// Each wave owns a 16-row strip; per 16-col tile one V_WMMA_F32_16X16X4_F32
// computes the 16x16 dot-product block. The loop tracks min SQUARED distance
// (sqrt is monotonic -> min commutes with sqrt); sqrt + clamp happen once per
// row in the epilogue, via raw v_sqrt_f32.
// ---------------------------------------------------------------------------
__global__ __launch_bounds__(256)
void chamfer_rowmin_kernel(const float* __restrict__ pred,
                           const float* __restrict__ tgt,
                           float* __restrict__ partials) {
  const int bs  = blockIdx.x;   // 0..39
  const int yt  = blockIdx.y;   // row tile group 0..15
  const int dir = blockIdx.z;   // 0 or 1

  const float* __restrict__ Prow = (dir == 0) ? pred : tgt;
  const float* __restrict__ Gcol = (dir == 0) ? tgt  : pred;
  const size_t base = (size_t)bs * N * 3;

  const int tid  = threadIdx.x;
  const int wave = tid >> 5;
  const int lane = tid & 31;
  const int hi   = lane >> 4;   // which 16-lane half of the wave
  const int l16  = lane & 15;

  const int rowBase = yt * ROWS_PER_WG + wave * ROWS_PER_WAVE;

  // A operand: 16x4 f32 (K=3 padded to 4).
  // Lanes 0-15: VGPR0=K0(x), VGPR1=K1(y); lanes 16-31: VGPR0=K2(z), VGPR1=K3(0)
  const float* pa = Prow + base + (size_t)(rowBase + l16) * 3;
  v2f a;
  a.x = hi ? pa[2] : pa[0];
  a.y = hi ? 0.0f  : pa[1];

  // ||p||^2 for the 8 rows this lane's C registers cover (M = r + 8*hi)
  float pn[8];
#pragma unroll
  for (int r = 0; r < 8; ++r) {
    const float* q = Prow + base + (size_t)(rowBase + hi * 8 + r) * 3;
    pn[r] = q[0] * q[0] + q[1] * q[1] + q[2] * q[2];
  }

  float rmin2[8];   // running min of squared distance
#pragma unroll
  for (int r = 0; r < 8; ++r) rmin2[r] = 3.0e38f;

  for (int jt = 0; jt < COL_TILES; ++jt) {
    const float* gb = Gcol + base + (size_t)(jt * 16 + l16) * 3;
    const float gx = gb[0], gy = gb[1], gz = gb[2];
    v2f b;
    b.x = hi ? gz   : gx;
    b.y = hi ? 0.0f : gy;
    const float gn = gx * gx + gy * gy + gz * gz;

    v8f c = {};
    // D = A(16x4) x B(4x16) + 0 : one 16x16 dot-product tile per wave
    c = __builtin_amdgcn_wmma_f32_16x16x4_f32(
        /*neg_a=*/false, a, /*neg_b=*/false, b,
        /*c_mod=*/(short)0, c, /*reuse_a=*/false, /*reuse_b=*/false);

#pragma unroll
    for (int r = 0; r < 8; ++r) {
      // d^2 = pn + gn - 2*dot   (no sqrt, no clamp in the hot loop)
      const float d2 = __builtin_fmaf(-2.0f, c[r], pn[r] + gn);
      rmin2[r] = fminf(rmin2[r], d2);
    }
  }

  // Row M's squared-min lives in the 16 lanes of its half (C layout:
  // VGPR r -> M = r + 8*hi, N = lane&15). Min-reduce within each half,
  // then sqrt once per row, accumulate, then merge halves.
  float rowSum = 0.0f;
#pragma unroll
  for (int r = 0; r < 8; ++r) {
    float v = rmin2[r];
    v = fminf(v, __shfl_xor(v, 1, 32));
    v = fminf(v, __shfl_xor(v, 2, 32));
    v = fminf(v, __shfl_xor(v, 4, 32));
    v = fminf(v, __shfl_xor(v, 8, 32));
    v = v > 0.0f ? v : 0.0f;                  // clamp commutes with min
    rowSum += __builtin_amdgcn_sqrtf(v);      // raw v_sqrt_f32, no fixup
  }
  rowSum += __shfl_xor(rowSum, 16, 32);   // sum over all 16 rows of the wave

  __shared__ float wsum[WAVES_PER_WG];
  if (lane == 0) wsum[wave] = rowSum;
  __syncthreads();
  if (tid == 0) {
    float s = 0.0f;
    for (int w = 0; w < WAVES_PER_WG; ++w) s += wsum[w];
    partials[((size_t)dir * BSN + bs) * ROW_TILES + yt] = s;  // deterministic
  }
}

// ---------------------------------------------------------------------------
// KL partial-sum kernel (5120 elements)
// ---------------------------------------------------------------------------
__global__ __launch_bounds__(256)
void kl_kernel(const float* __restrict__ pm, const float* __restrict__ pl,
               const float* __restrict__ qm, const float* __restrict__ ql,
               float* __restrict__ partials) {
  __shared__ float red[256];
  const int idx = blockIdx.x * 256 + threadIdx.x;
  float v = 0.0f;
  if (idx < KL_N) {
    const float a  = pl[idx];
    const float b2 = ql[idx];
    const float dm = qm[idx] - pm[idx];
    v = a - b2 + (__expf(b2) + dm * dm) * __expf(-a) - 1.0f;
  }
  red[threadIdx.x] = v;
  __syncthreads();
  for (int s = 128; s > 0; s >>= 1) {
    if (threadIdx.x < s) red[threadIdx.x] += red[threadIdx.x + s];
    __syncthreads();
  }
  if (threadIdx.x == 0) partials[blockIdx.x] = red[0];
}

// ---------------------------------------------------------------------------
// Temporal smoothness partial-sum kernel
// ---------------------------------------------------------------------------
__global__ __launch_bounds__(256)
void temporal_kernel(const float* __restrict__ pred,
                     float* __restrict__ partials) {
  __shared__ float red[256];
  const int idx = blockIdx.x * 256 + threadIdx.x;
  float v = 0.0f;
  if (idx < TMP_N) {
    const int n = idx % N;
    const int t = idx / N;      // 0 .. B*(S-1)-1
    const int s = t % (S - 1);
    const int b = t / (S - 1);
    const size_t i0 = (((size_t)b * S + s) * N + n) * 3;
    const size_t i1 = i0 + (size_t)N * 3;
    const float dx = pred[i1]     - pred[i0];
    const float dy = pred[i1 + 1] - pred[i0 + 1];
    const float dz = pred[i1 + 2] - pred[i0 + 2];
    v = __builtin_amdgcn_sqrtf(dx * dx + dy * dy + dz * dz);
  }
  red[threadIdx.x] = v;
  __syncthreads();
  for (int s2 = 128; s2 > 0; s2 >>= 1) {
    if (threadIdx.x < s2) red[threadIdx.x] += red[threadIdx.x + s2];
    __syncthreads();
  }
  if (threadIdx.x == 0) partials[blockIdx.x] = red[0];
}

// ---------------------------------------------------------------------------
// Finalize: sequential (deterministic) sum of partials, compute outputs
// ---------------------------------------------------------------------------
__global__ void finalize_kernel(const float* __restrict__ reconP,
                                const float* __restrict__ klP,
                                const float* __restrict__ tP,
                                float* __restrict__ out) {
  if (threadIdx.x == 0 && blockIdx.x == 0) {
    float rs = 0.0f;
    for (int i = 0; i < RECON_PARTIALS; ++i) rs += reconP[i];
    float ks = 0.0f;
    for (int i = 0; i < KL_BLK; ++i) ks += klP[i];
    float ts = 0.0f;
    for (int i = 0; i < TMP_BLK; ++i) ts += tP[i];

    const float recon    = rs / ((float)BSN * (float)N);
    const float kl       = 0.5f * ks / (float)BSN;
    const float temporal = ts / (float)TMP_N;
    const float total    = recon + kl + 0.1f * temporal;  // +0.1*uncertainty(=0)

    out[0] = total;
    out[1] = recon;
    out[2] = kl;
    out[3] = temporal;
    out[4] = 0.0f;   // uncertainty
  }
}

// ---------------------------------------------------------------------------
extern "C" void kernel_launch(void* const* d_in, const int* in_sizes, int n_in,
                              void* d_out, int out_size, void* d_ws, size_t ws_size,
                              hipStream_t stream) {
  (void)in_sizes; (void)n_in; (void)out_size; (void)ws_size;

  const float* pred = (const float*)d_in[0];
  const float* tgt  = (const float*)d_in[1];
  const float* pm   = (const float*)d_in[2];   // prior_mean
  const float* pl   = (const float*)d_in[3];   // prior_log_var
  const float* qm   = (const float*)d_in[4];   // posterior_mean
  const float* ql   = (const float*)d_in[5];   // posterior_log_var
  float* out = (float*)d_out;

  float* reconP = (float*)d_ws;                 // 1280 floats
  float* klP    = reconP + RECON_PARTIALS;      // 20 floats
  float* tP     = klP + KL_BLK;                 // 288 floats

  dim3 gridC(BSN, ROW_TILES, 2), blockC(256);
  chamfer_rowmin_kernel<<<gridC, blockC, 0, stream>>>(pred, tgt, reconP);
  kl_kernel<<<KL_BLK, 256, 0, stream>>>(pm, pl, qm, ql, klP);
  temporal_kernel<<<TMP_BLK, 256, 0, stream>>>(pred, tP);
  finalize_kernel<<<1, 32, 0, stream>>>(reconP, klP, tP, out);
}